// GraphConvolution_37357625541289
// MI455X (gfx1250) — compile-verified
//
#include <hip/hip_runtime.h>

// Graph convolution: out = relu(sum_s SpMM(A_s, x @ K_s) + bias)
// B=16, N=6890, F=O=256, S=2, E=55120
typedef __attribute__((ext_vector_type(2))) float v2f;
typedef __attribute__((ext_vector_type(8))) float v8f;

#define GC_B 16
#define GC_N 6890
#define GC_F 256
#define GC_O 256
#define GC_S 2
#define GC_E 55120
#define GC_M (GC_B * GC_N)          // 110240 flattened rows of x: [B*N, F]

#define BM 128                      // block M tile
#define BN 128                      // block N tile (2 N-blocks over O=256)
#define BK 32                       // K tile
#define XS_STRIDE (BK + 4)          // 36: A-fragment column reads conflict-free (36*lo mod 64 all distinct, mult. of 4)
#define KP (BK / 2)                 // 16 k-pairs
#define KS2 (2 * BN + 32)           // 288 ≡ 32 (mod 64): half-waves (adjacent k-pairs) hit disjoint bank halves

// ---------------------------------------------------------------- zero out (b128 stores)
__global__ __launch_bounds__(256) void gc_zero4(float4* __restrict__ p, size_t n4) {
    size_t i = (size_t)blockIdx.x * blockDim.x + threadIdx.x;
    if (i < n4) p[i] = make_float4(0.f, 0.f, 0.f, 0.f);
}

// ---------------------------------------------------------------- GEMM: P[M,O] = X[M,F] @ Kw[F,O]
__global__ __launch_bounds__(256) void gc_gemm(const float* __restrict__ X,
                                               const float* __restrict__ Kw,
                                               float* __restrict__ P) {
    __shared__ float xs[BM * XS_STRIDE];   // 128 x 32 (+pad)          = 18 KB
    __shared__ float ks[KP * KS2];         // 16 k-pairs x 128 n (x2)  = 18 KB, pair-interleaved

    const int tid  = threadIdx.x;
    const int wave = tid >> 5;             // 0..7 (wave32)
    const int lane = tid & 31;
    const int lo   = lane & 15;
    const int hi   = lane >> 4;            // 0 or 1
    const int m0   = blockIdx.x * BM;
    const int n0   = blockIdx.y * BN;
    const int wm   = (wave & 3) * 32;      // wave's 32-row strip
    const int wn   = (wave >> 2) * 64;     // wave's 64-col strip

    v8f acc[2][4];
#pragma unroll
    for (int mi = 0; mi < 2; ++mi)
#pragma unroll
        for (int t = 0; t < 4; ++t) acc[mi][t] = (v8f){};

    for (int k0 = 0; k0 < GC_F; k0 += BK) {
        // --- stage X tile: 128 rows x 32 floats, one b128 per thread x4 (coalesced)
#pragma unroll
        for (int i = 0; i < 4; ++i) {
            int idx = tid + i * 256;               // 0..1023
            int r   = idx >> 3;                    // tile row 0..127
            int c   = (idx & 7) << 2;              // tile col 0..28
            int gr  = m0 + r;
            if (gr >= GC_M) gr = GC_M - 1;         // clamp (edge tile; stores are guarded)
            const float4 v = *reinterpret_cast<const float4*>(X + (size_t)gr * GC_F + k0 + c);
            float* dst = &xs[r * XS_STRIDE + c];
            dst[0] = v.x; dst[1] = v.y; dst[2] = v.z; dst[3] = v.w;
        }
        // --- stage K tile 32x128, PAIR-INTERLEAVED: ks[p][2n+bit] = Kw[k0+2p+bit][n0+n]
        //     => every B fragment (rows 2p,2p+1 at one column) is one contiguous b64 in LDS
#pragma unroll
        for (int i = 0; i < 4; ++i) {
            int idx = tid + i * 256;               // 0..1023
            int np  = idx & 63;                    // column pair 0..63
            int p   = idx >> 6;                    // k-pair 0..15
            const float* g0 = Kw + (size_t)(k0 + 2 * p) * GC_O + n0 + 2 * np;
            const v2f r0 = *reinterpret_cast<const v2f*>(g0);          // row 2p,   cols {2np,2np+1}
            const v2f r1 = *reinterpret_cast<const v2f*>(g0 + GC_O);   // row 2p+1, cols {2np,2np+1}
            float4 w; w.x = r0.x; w.y = r1.x; w.z = r0.y; w.w = r1.y;  // interleave
            *reinterpret_cast<float4*>(&ks[p * KS2 + 4 * np]) = w;     // 16B-aligned b128 store
        }
        __syncthreads();

        // --- 8 K-steps; per step: 2 A-frag b64 + 4 B-frag b64 feed 8 v_wmma_f32_16x16x4_f32
#pragma unroll
        for (int kk = 0; kk < BK; kk += 4) {
            const int p = (kk >> 1) + hi;          // k-pair for this half-wave (rows kk+2hi, kk+2hi+1)
            v2f a[2];
            a[0] = *reinterpret_cast<const v2f*>(&xs[(wm + lo)      * XS_STRIDE + kk + 2 * hi]);
            a[1] = *reinterpret_cast<const v2f*>(&xs[(wm + 16 + lo) * XS_STRIDE + kk + 2 * hi]);
#pragma unroll
            for (int t = 0; t < 4; ++t) {
                const v2f b = *reinterpret_cast<const v2f*>(&ks[p * KS2 + 2 * (wn + t * 16 + lo)]);
                acc[0][t] = __builtin_amdgcn_wmma_f32_16x16x4_f32(
                    false, a[0], false, b, (short)0, acc[0][t], false, false);
                acc[1][t] = __builtin_amdgcn_wmma_f32_16x16x4_f32(
                    false, a[1], false, b, (short)0, acc[1][t], false, false);
            }
        }
        __syncthreads();
    }

    // --- store C: VGPR j -> row M = j + 8*hi, col N = lo (16x16 f32 C layout)
#pragma unroll
    for (int mi = 0; mi < 2; ++mi)
#pragma unroll
        for (int j = 0; j < 8; ++j) {
            const int gr = m0 + wm + mi * 16 + j + 8 * hi;
            if (gr < GC_M) {
                float* row = P + (size_t)gr * GC_O + n0 + wn;
#pragma unroll
                for (int t = 0; t < 4; ++t) row[t * 16 + lo] = acc[mi][t][j];
            }
        }
}

// ---------------------------------------------------------------- SpMM: out[b,rows[e],:] += vals[e]*P[b,cols[e],:]
__global__ __launch_bounds__(256) void gc_spmm(const int* __restrict__ rows,
                                               const int* __restrict__ cols,
                                               const float* __restrict__ vals,
                                               const float* __restrict__ P,
                                               float* __restrict__ out) {
    const int e = blockIdx.x;              // uniform -> scalar loads
    const int r = rows[e];
    const int c = cols[e];
    const float v = vals[e];
    const int o = threadIdx.x;             // 0..255 output channel
#pragma unroll
    for (int b = 0; b < GC_B; ++b) {
        const size_t src = ((size_t)b * GC_N + c) * GC_O + o;
        const size_t dst = ((size_t)b * GC_N + r) * GC_O + o;
        unsafeAtomicAdd(out + dst, v * P[src]);   // native global_atomic_add_f32; pre is L2-resident
    }
}

// ---------------------------------------------------------------- bias + relu (b128 path)
__global__ __launch_bounds__(256) void gc_bias_relu4(float4* __restrict__ out,
                                                     const float4* __restrict__ bias4,
                                                     size_t n4) {
    size_t i = (size_t)blockIdx.x * blockDim.x + threadIdx.x;
    if (i < n4) {
        const float4 b = bias4[i & (GC_O / 4 - 1)];   // O=256 -> 64 float4 per row
        float4 v = out[i];
        v.x = fmaxf(v.x + b.x, 0.f);
        v.y = fmaxf(v.y + b.y, 0.f);
        v.z = fmaxf(v.z + b.z, 0.f);
        v.w = fmaxf(v.w + b.w, 0.f);
        out[i] = v;
    }
}

// ---------------------------------------------------------------- launcher
extern "C" void kernel_launch(void* const* d_in, const int* in_sizes, int n_in,
                              void* d_out, int out_size, void* d_ws, size_t ws_size,
                              hipStream_t stream) {
    const float* x       = (const float*)d_in[0];   // [B,N,F]
    const int*   rows    = (const int*)  d_in[1];   // [S,E]
    const int*   cols    = (const int*)  d_in[2];   // [S,E]
    const float* vals    = (const float*)d_in[3];   // [S,E]
    const float* kernels = (const float*)d_in[4];   // [S,F,O]
    const float* bias    = (const float*)d_in[5];   // [O]
    float* out = (float*)d_out;                     // [B,N,O]
    float* pre = (float*)d_ws;                      // [B*N, O] scratch, 113 MB (fits in 192 MB L2)

    const size_t total = (size_t)GC_M * GC_O;       // 28,221,440 (divisible by 1024)
    const size_t n4    = total / 4;

    gc_zero4<<<dim3((unsigned)(n4 / 256)), 256, 0, stream>>>((float4*)out, n4);

    const dim3 ggrid((GC_M + BM - 1) / BM, GC_O / BN);   // 862 x 2
    for (int s = 0; s < GC_S; ++s) {
        gc_gemm<<<ggrid, 256, 0, stream>>>(x, kernels + (size_t)s * GC_F * GC_O, pre);
        gc_spmm<<<GC_E, 256, 0, stream>>>(rows + (size_t)s * GC_E,
                                          cols + (size_t)s * GC_E,
                                          vals + (size_t)s * GC_E,
                                          pre, out);
    }

    gc_bias_relu4<<<dim3((unsigned)(n4 / 256)), 256, 0, stream>>>(
        (float4*)out, (const float4*)bias, n4);
}